// LNon_36816459662158
// MI455X (gfx1250) — compile-verified
//
#include <hip/hip_runtime.h>

// MI455X / gfx1250 elementwise streaming kernel.
// Reference is a pure pointwise map (the (1,1,1) "matmuls" are scalar mults),
// so no WMMA applies; the CDNA5 path exercised here is the async
// global->LDS streaming pipeline (ASYNCcnt) + explicit ds_load_b128 reads
// + NT 128-bit stores.

#define THREADS    256
#define DEPTH      3                 // async pipeline depth (triple buffer)
#define TILE_ELEMS 1024              // THREADS * 4 floats
#define TILE_BYTES 4096

typedef float f32x4 __attribute__((ext_vector_type(4)));

struct LNonConsts {
  float t0, ts0, ts1, ts2, ts3;      // theta table pre-scaled by 1/(2*pi)
  float v0, vs0, vs1, vs2, vs3;      // velocity table pre-scaled by STEP_LENGTH
  float cscale, sscale;
};

__device__ __forceinline__ LNonConsts make_consts(const float* __restrict__ th,
                                                  const float* __restrict__ ve,
                                                  const float* __restrict__ ct,
                                                  const float* __restrict__ st) {
  const float inv2pi = 0.15915494309189535f;   // fold into theta: v_sin/v_cos take x/(2pi)
  const float step   = 0.33333f;               // STEP_LENGTH folded into velocity (lerp is linear)
  LNonConsts k;
  float t0 = th[0] * inv2pi, t1 = th[1] * inv2pi, t2 = th[2] * inv2pi,
        t3 = th[3] * inv2pi, t4 = th[4] * inv2pi;
  float w0 = ve[0] * step, w1 = ve[1] * step, w2 = ve[2] * step,
        w3 = ve[3] * step, w4 = ve[4] * step;
  k.t0 = t0; k.ts0 = t1 - t0; k.ts1 = t2 - t1; k.ts2 = t3 - t2; k.ts3 = t4 - t3;
  k.v0 = w0; k.vs0 = w1 - w0; k.vs1 = w2 - w1; k.vs2 = w3 - w2; k.vs3 = w4 - w3;
  k.cscale = ct[0]; k.sscale = st[0];
  return k;
}

// 5-point table lerp (with the reference's clamp + last-segment extrapolation)
// as a segment sum: T0 + sum_k s_k * clamp(i-k, 0, cap_k). Each clamp = v_med3_f32.
// cap3 = 2 reproduces the double-decrement extrapolation for idx in [4,5).
__device__ __forceinline__ float seg_eval(float i, float b, float s0, float s1,
                                          float s2, float s3) {
  float r = fmaf(s0, __builtin_amdgcn_fmed3f(i,        0.0f, 1.0f), b);
  r = fmaf(s1, __builtin_amdgcn_fmed3f(i - 1.0f, 0.0f, 1.0f), r);
  r = fmaf(s2, __builtin_amdgcn_fmed3f(i - 2.0f, 0.0f, 1.0f), r);
  r = fmaf(s3, __builtin_amdgcn_fmed3f(i - 3.0f, 0.0f, 2.0f), r);
  return r;
}

__device__ __forceinline__ float lnon_point(float xv, const LNonConsts& k) {
  float d = xv * k.cscale;
#pragma unroll
  for (int s = 0; s < 3; ++s) {
    // it = 5 * sigmoid(d)  via v_exp_f32 (2^x) + v_rcp_f32
    float e  = __builtin_amdgcn_exp2f(d * -1.44269504f);
    float it = 5.0f * __builtin_amdgcn_rcpf(1.0f + e);
    // iv = 5 * 0.5*(1+erf(d))  via Abramowitz-Stegun 7.1.26 (|err|<1.5e-7)
    float ad = fabsf(d);
    float t  = __builtin_amdgcn_rcpf(fmaf(0.3275911f, ad, 1.0f));
    float p  = fmaf(fmaf(fmaf(fmaf(1.061405429f, t, -1.453152027f), t,
                     1.421413741f), t, -0.284496736f), t, 0.254829592f);
    float eh = __builtin_amdgcn_exp2f(fmaf(-1.44269504f, d * d, -1.0f)); // 0.5*exp(-d*d)
    float g  = 5.0f * (p * t) * eh;                 // 2.5*(1 - erf(|d|))
    float iv = (d >= 0.0f) ? (5.0f - g) : g;

    float th = seg_eval(it, k.t0, k.ts0, k.ts1, k.ts2, k.ts3); // already /(2pi)
    float ds = seg_eval(iv, k.v0, k.vs0, k.vs1, k.vs2, k.vs3); // already *STEP_LENGTH
    float sn = __builtin_amdgcn_sinf(th);
    float cs = __builtin_amdgcn_cosf(th);
    d = fmaf(d, fmaf(ds, sn, 1.0f), ds * cs);       // d*(1+dy)+dx
  }
  return d * k.sscale;
}

// Streaming kernel: triple-buffered async global->LDS pipeline.
// Each wave async-loads only its own 512B slice of each 4KB tile, so the
// per-wave ASYNCcnt wait is the only synchronization needed (no barriers).
__global__ __launch_bounds__(THREADS) void lnon_stream_kernel(
    const float* __restrict__ x, const float* __restrict__ theta,
    const float* __restrict__ vel, const float* __restrict__ ct,
    const float* __restrict__ st, float* __restrict__ out,
    int q, int r) {
  __shared__ __align__(16) float lds_buf[DEPTH * TILE_ELEMS];  // 12 KB
  const int tid = threadIdx.x;
  const int bid = blockIdx.x;
  const int tiles = q + (bid < r ? 1 : 0);
  if (tiles == 0) return;
  const long long first = (long long)bid * q + (bid < r ? bid : r);

  const LNonConsts k = make_consts(theta, vel, ct, st);

  const float* xbase = x + first * TILE_ELEMS;
  float*       obase = out + first * TILE_ELEMS;
  const unsigned lane16 = (unsigned)tid * 16u;
  // LDS byte offset of this lane's slot in stage 0. Derived from the real
  // pointer (ptrtoint => lds_buf is captured/kept alive); low 32 bits of a
  // generic LDS pointer are the DS byte offset.
  const unsigned lds_lane0 = (unsigned)(unsigned long long)(&lds_buf[tid * 4]);

  // Issue one async 16B load per lane for (stage, tile t).
#define ISSUE_ASYNC(stage_, t_)                                                \
  do {                                                                         \
    unsigned lds_addr_ = lds_lane0 + (unsigned)(stage_) * TILE_BYTES;          \
    unsigned goff_ = (unsigned)(t_) * TILE_BYTES + lane16;                     \
    asm volatile("global_load_async_to_lds_b128 %0, %1, %2 th:TH_LOAD_NT"      \
                 :: "v"(lds_addr_), "v"(goff_), "s"(xbase)                     \
                 : "memory");                                                  \
  } while (0)

  // Explicit ds_load_b128 read (opaque to the optimizer) + compute + NT store.
#define CONSUME_TILE(stage_, i_)                                               \
  do {                                                                         \
    unsigned lds_addr_ = lds_lane0 + (unsigned)(stage_) * TILE_BYTES;          \
    f32x4 v_;                                                                  \
    asm volatile("ds_load_b128 %0, %1\n\ts_wait_dscnt 0"                       \
                 : "=v"(v_) : "v"(lds_addr_) : "memory");                      \
    f32x4 o_;                                                                  \
    o_.x = lnon_point(v_.x, k); o_.y = lnon_point(v_.y, k);                    \
    o_.z = lnon_point(v_.z, k); o_.w = lnon_point(v_.w, k);                    \
    __builtin_nontemporal_store(                                               \
        o_, (f32x4*)(obase + (long long)(i_) * TILE_ELEMS + tid * 4));         \
  } while (0)

  // Prologue: always fill DEPTH slots (duplicate the last real tile into
  // unused stages so the ASYNCcnt bookkeeping stays compile-time constant).
#pragma unroll
  for (int p = 0; p < DEPTH; ++p) {
    const int t = (p < tiles) ? p : (tiles - 1);
    ISSUE_ASYNC(p, t);
  }

  int i = 0;
  int stage = 0;
  // Steady state: oldest tile done when ASYNCcnt <= DEPTH-1 (in-order completion).
  for (; i + DEPTH < tiles; ++i) {
    asm volatile("s_wait_asynccnt %0" :: "n"(DEPTH - 1) : "memory");
    CONSUME_TILE(stage, i);
    ISSUE_ASYNC(stage, i + DEPTH);   // refill the just-consumed stage
    stage = (stage + 1 == DEPTH) ? 0 : stage + 1;
  }
  // Drain: wait levels count down at compile time under full unroll.
#pragma unroll
  for (int j = 0; j < DEPTH; ++j) {
    if (i >= tiles) break;
    asm volatile("s_wait_asynccnt %0" :: "n"(DEPTH - 1 - j) : "memory");
    CONSUME_TILE(stage, i);
    stage = (stage + 1 == DEPTH) ? 0 : stage + 1;
    ++i;
  }
#undef ISSUE_ASYNC
#undef CONSUME_TILE
}

// Guarded scalar tail for element counts not divisible by TILE_ELEMS
// (empty for the reference's 16*4096*1024 shape).
__global__ void lnon_tail_kernel(const float* __restrict__ x,
                                 const float* __restrict__ theta,
                                 const float* __restrict__ vel,
                                 const float* __restrict__ ct,
                                 const float* __restrict__ st,
                                 float* __restrict__ out,
                                 long long start, long long n) {
  long long idx = start + (long long)blockIdx.x * blockDim.x + threadIdx.x;
  if (idx < n) {
    const LNonConsts k = make_consts(theta, vel, ct, st);
    out[idx] = lnon_point(x[idx], k);
  }
}

extern "C" void kernel_launch(void* const* d_in, const int* in_sizes, int n_in,
                              void* d_out, int out_size, void* d_ws, size_t ws_size,
                              hipStream_t stream) {
  (void)n_in; (void)d_ws; (void)ws_size; (void)out_size;
  const float* x     = (const float*)d_in[0];
  const float* theta = (const float*)d_in[1];
  const float* vel   = (const float*)d_in[2];
  const float* ct    = (const float*)d_in[3];
  const float* st    = (const float*)d_in[4];
  float* out = (float*)d_out;

  const long long n = (long long)in_sizes[0];
  const int ntiles = (int)(n / TILE_ELEMS);
  if (ntiles > 0) {
    const int grid = ntiles < 8192 ? ntiles : 8192;  // ~8 contiguous 4KB tiles/block
    const int q = ntiles / grid;
    const int r = ntiles % grid;
    lnon_stream_kernel<<<grid, THREADS, 0, stream>>>(x, theta, vel, ct, st, out, q, r);
  }
  const long long start = (long long)ntiles * TILE_ELEMS;
  if (start < n) {
    const long long remn = n - start;
    const int g2 = (int)((remn + THREADS - 1) / THREADS);
    lnon_tail_kernel<<<g2, THREADS, 0, stream>>>(x, theta, vel, ct, st, out, start, n);
  }
}